// MedianFilter_13168369729578
// MI455X (gfx1250) — compile-verified
//
#include <hip/hip_runtime.h>
#include <stdint.h>

#define IMG_H 512
#define IMG_W 512
#define TILE 32
#define HALO 2
#define IN_T (TILE + 2 * HALO)   // 36

typedef uint32_t u32x4 __attribute__((ext_vector_type(4)));
typedef uint32_t u32x8 __attribute__((ext_vector_type(8)));

// reflect (mirror without edge repeat), valid for i in [-(n-1), 2n-2)
__device__ __forceinline__ int refl(int i, int n) {
  i = (i < 0) ? -i : i;
  i = (i >= n) ? (2 * n - 2 - i) : i;
  return i;
}

__device__ __forceinline__ void cswap(float &a, float &b) {
  float lo = fminf(a, b);
  float hi = fmaxf(a, b);
  a = lo;
  b = hi;
}

// Push min of w[0..C-1] to w[0] and max to w[C-1] (multiset-preserving).
// Pairwise + tree: 3C/2-2 CE (even C), 3(C-1)/2 CE (odd C).
template <int C>
__device__ __forceinline__ void minmax_pass(float *w) {
  if constexpr ((C & 1) == 0) {
#pragma unroll
    for (int i = 0; i < C / 2; ++i) cswap(w[2 * i], w[2 * i + 1]);
#pragma unroll
    for (int i = 1; i < C / 2; ++i) cswap(w[0], w[2 * i]);       // min tree (lows)
#pragma unroll
    for (int i = 0; i < C / 2 - 1; ++i) cswap(w[2 * i + 1], w[C - 1]);  // max tree (highs)
  } else {
#pragma unroll
    for (int i = 0; i < (C - 1) / 2; ++i) cswap(w[2 * i], w[2 * i + 1]);
#pragma unroll
    for (int i = 1; i < (C - 1) / 2; ++i) cswap(w[0], w[2 * i]); // min tree (lows)
    cswap(w[0], w[C - 1]);                                       // fold in unpaired elem
#pragma unroll
    for (int i = 0; i < (C - 1) / 2; ++i) cswap(w[2 * i + 1], w[C - 1]);  // max tree
  }
}

// Forgetful selection: median (13th of 25). Working set starts at 25-13+2 = 14.
__device__ __forceinline__ float median25(const float *v) {
  float w[14];
#pragma unroll
  for (int i = 0; i < 14; ++i) w[i] = v[i];
  minmax_pass<14>(w); w[0] = v[14];
  minmax_pass<13>(w); w[0] = v[15];
  minmax_pass<12>(w); w[0] = v[16];
  minmax_pass<11>(w); w[0] = v[17];
  minmax_pass<10>(w); w[0] = v[18];
  minmax_pass<9>(w);  w[0] = v[19];
  minmax_pass<8>(w);  w[0] = v[20];
  minmax_pass<7>(w);  w[0] = v[21];
  minmax_pass<6>(w);  w[0] = v[22];
  minmax_pass<5>(w);  w[0] = v[23];
  minmax_pass<4>(w);  w[0] = v[24];
  minmax_pass<3>(w);  // w[0]=min, w[2]=max -> w[1] is the median
  return w[1];
}

__global__ __launch_bounds__(256) void median5_kernel(
    const float *__restrict__ img, float *__restrict__ out) {
  __shared__ float tile[IN_T * IN_T];

  const int plane = blockIdx.z;  // b*3 + c
  const int x0 = blockIdx.x * TILE;
  const int y0 = blockIdx.y * TILE;
  const float *src = img + (size_t)plane * IMG_H * IMG_W;

  const int tid = threadIdx.x;
  const uint32_t ldsBase = (uint32_t)(uintptr_t)(&tile[0]);

  const bool interior = (x0 >= HALO) && (x0 + TILE + HALO <= IMG_W) &&
                        (y0 >= HALO) && (y0 + TILE + HALO <= IMG_H);

  if (interior) {
    // ---- Interior tile: single Tensor Data Mover 2D tile load -> LDS ----
    if (tid < 32) {  // one wave issues the TDM op (EXEC is ignored by TDM)
      const uint64_t gaddr =
          (uint64_t)(uintptr_t)(src + (size_t)(y0 - HALO) * IMG_W + (x0 - HALO));
      // D# group 0: count=1 | lds_addr | global_addr[56:0] | type=2
      const u32x4 g0 = {1u, ldsBase, (uint32_t)gaddr,
                        (uint32_t)(gaddr >> 32) | 0x80000000u};
      // D# group 1: data_size=4B; tensor_dim0/1=36; tile_dim0/1=36;
      //             tensor_dim0_stride=512 elements
      const u32x8 g1 = {0x00020000u,          // workgroup_mask=0, data_size=2 (4B)
                        (uint32_t)IN_T << 16, // tensor_dim0[15:0]
                        (uint32_t)IN_T << 16, // tensor_dim0[31:16]=0, tensor_dim1[15:0]
                        (uint32_t)IN_T << 16, // tensor_dim1[31:16]=0, tile_dim0
                        (uint32_t)IN_T,       // tile_dim1 (tile_dim2=0)
                        (uint32_t)IMG_W,      // tensor_dim0_stride[31:0]
                        0u, 0u};
      asm volatile("tensor_load_to_lds %0, %1" :: "s"(g0), "s"(g1) : "memory");
      __builtin_amdgcn_s_wait_tensorcnt(0);
    }
  } else {
    // ---- Border tile: per-lane reflected async global->LDS loads ----
#pragma unroll
    for (int it = 0; it < 6; ++it) {
      const int e = tid + it * 256;
      if (e < IN_T * IN_T) {
        const int ly = e / IN_T;
        const int lx = e - ly * IN_T;
        const int gy = refl(y0 - HALO + ly, IMG_H);
        const int gx = refl(x0 - HALO + lx, IMG_W);
        const uint32_t goff = (uint32_t)((gy * IMG_W + gx) * 4);
        const uint32_t laddr = ldsBase + (uint32_t)((ly * IN_T + lx) * 4);
        asm volatile("global_load_async_to_lds_b32 %0, %1, %2"
                     :
                     : "v"(laddr), "v"(goff), "s"(src)
                     : "memory");
      }
    }
    asm volatile("s_wait_asynccnt 0" ::: "memory");
  }
  __syncthreads();

  // ---- Compute: each thread does a 4-tall column of outputs ----
  const int tx = tid & 31;         // output column within tile
  const int tyb = (tid >> 5) * 4;  // base output row within tile
  float *outp = out + (size_t)plane * IMG_H * IMG_W;

#pragma unroll
  for (int r = 0; r < 4; ++r) {
    const int ty = tyb + r;
    float v[25];
#pragma unroll
    for (int dy = 0; dy < 5; ++dy) {
#pragma unroll
      for (int dx = 0; dx < 5; ++dx) {
        v[dy * 5 + dx] = tile[(ty + dy) * IN_T + (tx + dx)];
      }
    }
    outp[(size_t)(y0 + ty) * IMG_W + (x0 + tx)] = median25(v);
  }
}

extern "C" void kernel_launch(void *const *d_in, const int *in_sizes, int n_in,
                              void *d_out, int out_size, void *d_ws,
                              size_t ws_size, hipStream_t stream) {
  (void)n_in;
  (void)d_ws;
  (void)ws_size;
  (void)out_size;
  const float *img = (const float *)d_in[0];
  float *out = (float *)d_out;
  const int planes = in_sizes[0] / (IMG_H * IMG_W);  // 4*3 = 12
  dim3 grid(IMG_W / TILE, IMG_H / TILE, planes);
  median5_kernel<<<grid, dim3(256), 0, stream>>>(img, out);
}